// Upsample_1803886264376
// MI455X (gfx1250) — compile-verified
//
#include <hip/hip_runtime.h>

// upfirdn2d upsample x2 with [1,3,3,1] separable kernel, pad=(2,1).
// Polyphase: each output pixel is a (1,3)/(3,1)-weighted 2x2 input blend / 16.
//
// Input : (8,256,128,128) f32 (128 MB), Output: (8,256,256,256) f32 (512 MB)
// -> pure bandwidth problem (~1.7 FLOP/byte). Strategy:
//   * full-width input tiles (16 rows x 128 cols) staged to LDS with the
//     gfx1250 Tensor Data Mover (tensor_load_to_lds): one 2D descriptor per
//     wave slice, hardware DMA does the strided walk + LDS row padding
//   * left/right halo = image boundary -> pre-zeroed LDS pad columns
//   * compute 32x256 output tile from LDS, two 16B non-temporal stores/thread/pass

#define IH 128
#define IW 128
#define OW 256
#define TROWS 16          // input rows per tile
#define SROWS 18          // + top/bottom halo rows
#define LPAD 4            // left pad (floats); keeps DMA LDS dest 16B aligned
#define SSTRIDE 136       // LPAD + 128 data + 4 right pad; = 128 dw + 8 dw pad

typedef float v4f __attribute__((ext_vector_type(4)));
typedef unsigned int v4u __attribute__((ext_vector_type(4)));
typedef int v8i __attribute__((ext_vector_type(8)));
typedef int v4i __attribute__((ext_vector_type(4)));

__global__ __launch_bounds__(256) void upsample2x_1331_kernel(
    const float* __restrict__ in, float* __restrict__ out)
{
    __shared__ float smem[SROWS * SSTRIDE];   // 9792 B

    const int tile = blockIdx.x;         // row tile 0..7
    const int ch   = blockIdx.y;         // fused (batch, channel) 0..2047
    const int tid  = threadIdx.x;

    const float* __restrict__ src = in  + (size_t)ch * (IH * IW);
    float* __restrict__       dst = out + (size_t)ch * (2 * IH * OW);

    const int iy0 = tile * TROWS;

    // ---- Zero the pad columns (image left/right boundary -> zero halo).
    if (tid < SROWS * 8) {
        const int r = tid >> 3;
        const int p = tid & 7;
        const int c = (p < 4) ? p : (128 + p);
        smem[r * SSTRIDE + c] = 0.0f;
    }
    // ---- Zero out-of-image halo rows (top of first tile, bottom of last).
    if (iy0 == 0 && tid < 32)
        *(v4f*)&smem[0 * SSTRIDE + LPAD + 4 * tid] = (v4f){0.f, 0.f, 0.f, 0.f};
    if (iy0 + TROWS >= IH && tid < 32)
        *(v4f*)&smem[(SROWS - 1) * SSTRIDE + LPAD + 4 * tid] = (v4f){0.f, 0.f, 0.f, 0.f};

    // ---- Stage input rows iy0-1 .. iy0+16 (clipped to image) into LDS.
    const int r0    = (iy0 == 0) ? 1 : 0;                  // first LDS row DMAd
    const int rend  = (iy0 + TROWS >= IH) ? (SROWS - 1) : SROWS;
    const int nrows = rend - r0;                           // 16..18

#if __has_builtin(__builtin_amdgcn_tensor_load_to_lds)
    // Tensor Data Mover path: each of the 8 waves DMAs its 2-3 row slice with
    // one 2D descriptor; TDM applies the 8-DWORD LDS row padding in hardware.
    {
        const int wave = tid >> 5;                         // wave-uniform
        const int rs = r0 + (nrows * wave) / 8;            // slice start row
        const int rc = r0 + (nrows * (wave + 1)) / 8 - rs; // slice row count

        const unsigned long long gaddr =
            (unsigned long long)(size_t)(src + (iy0 - 1 + rs) * IW);
        const unsigned lds_addr = (unsigned)(size_t)&smem[rs * SSTRIDE + LPAD];

        v4u g0;
        g0.x = 0x1u;                                   // count=1, user mode
        g0.y = lds_addr;                               // lds_addr[31:0]
        g0.z = (unsigned)(gaddr & 0xFFFFFFFFull);      // global_addr[31:0]
        g0.w = (unsigned)((gaddr >> 32) & 0x01FFFFFFull) | (2u << 30); // type=2

        v8i g1;
        g1[0] = (int)((2u << 16)      // data_size = 4 bytes
                    | (1u << 20)      // pad_enable
                    | (6u << 22)      // pad_interval: every 128 DWORDs
                    | (7u << 25));    // pad_amount : 8 DWORDs (-> stride 136)
        g1[1] = (int)(128u << 16);    // tensor_dim0[15:0]=128 (in bits 79:48)
        g1[2] = (int)(((unsigned)rc) << 16); // dim0 hi=0 | tensor_dim1 lo = rc
        g1[3] = (int)(128u << 16);    // tensor_dim1 hi=0 | tile_dim0 = 128
        g1[4] = rc;                   // tile_dim1 = rc | tile_dim2 = 0
        g1[5] = 128;                  // tensor_dim0_stride[31:0] = 128 elements
        g1[6] = 0;                    // stride0 hi | tensor_dim1_stride lo
        g1[7] = 0;

        const v4i zero4 = {0, 0, 0, 0};                // 2D: groups 2/3 unused
        const v8i zero8 = {0, 0, 0, 0, 0, 0, 0, 0};    // extra group (clang-23 ABI)
        __builtin_amdgcn_tensor_load_to_lds(g0, g1, zero4, zero4, zero8, 0);
        __builtin_amdgcn_s_wait_tensorcnt(0);          // this wave's DMA landed
    }
#else
    // Fallback: per-lane gfx1250 async global->LDS DMA, 128-bit per lane.
    for (int idx = tid; idx < SROWS * (IW / 4); idx += 256) {
        const int r = idx >> 5;
        const int k = idx & 31;
        const int gy = iy0 - 1 + r;
        float* sp = &smem[r * SSTRIDE + LPAD + 4 * k];
        if ((unsigned)gy < (unsigned)IH) {
            const unsigned lds_addr = (unsigned)(size_t)sp;
            const unsigned long long ga =
                (unsigned long long)(size_t)(src + gy * IW + 4 * k);
            asm volatile("global_load_async_to_lds_b128 %0, %1, off"
                         :: "v"(lds_addr), "v"(ga) : "memory");
        }
    }
    asm volatile("s_wait_asynccnt 0" ::: "memory");
#endif
    __syncthreads();                                   // everyone's fills visible

    // ---- Compute: thread covers input col cells (n0, n0+1); per pass one input
    // row cell m -> 2x4 output block at (2(iy0+m), 2n0).
    const int tx = tid & 63;             // 64 column groups
    const int ry = tid >> 6;             // 0..3
    const int n0 = tx * 2;               // local input col cell, even, 0..126

    const float inv16 = 1.0f / 16.0f;

    #pragma unroll
    for (int pass = 0; pass < 4; ++pass) {
        const int m = ry + pass * 4;     // local input row cell 0..15

        // smem row m+0 == input row m-1; smem col (LPAD-1+n0) == input col n0-1
        const float* s0 = &smem[(m + 0) * SSTRIDE + (LPAD - 1) + n0];
        const float* s1 = s0 + SSTRIDE;
        const float* s2 = s1 + SSTRIDE;

        const float a00 = s0[0], a01 = s0[1], a02 = s0[2], a03 = s0[3];
        const float a10 = s1[0], a11 = s1[1], a12 = s1[2], a13 = s1[3];
        const float a20 = s2[0], a21 = s2[1], a22 = s2[2], a23 = s2[3];

        // Horizontal polyphase: even out col -> (1,3), odd -> (3,1)
        const float h00 = a00 + 3.f * a01, h01 = 3.f * a01 + a02;
        const float h02 = a01 + 3.f * a02, h03 = 3.f * a02 + a03;
        const float h10 = a10 + 3.f * a11, h11 = 3.f * a11 + a12;
        const float h12 = a11 + 3.f * a12, h13 = 3.f * a12 + a13;
        const float h20 = a20 + 3.f * a21, h21 = 3.f * a21 + a22;
        const float h22 = a21 + 3.f * a22, h23 = 3.f * a22 + a23;

        // Vertical polyphase: even out row -> (1,3), odd -> (3,1); fold /16.
        v4f o0, o1;
        o0.x = (h00 + 3.f * h10) * inv16;
        o0.y = (h01 + 3.f * h11) * inv16;
        o0.z = (h02 + 3.f * h12) * inv16;
        o0.w = (h03 + 3.f * h13) * inv16;
        o1.x = (3.f * h10 + h20) * inv16;
        o1.y = (3.f * h11 + h21) * inv16;
        o1.z = (3.f * h12 + h22) * inv16;
        o1.w = (3.f * h13 + h23) * inv16;

        const int oy = (iy0 + m) * 2;
        const int ox = n0 * 2;                       // multiple of 4 -> 16B aligned
        float* p0 = dst + (size_t)oy * OW + ox;
        float* p1 = p0 + OW;
        __builtin_nontemporal_store(o0, (v4f*)p0);   // write-once stream: NT hint
        __builtin_nontemporal_store(o1, (v4f*)p1);
    }
}

extern "C" void kernel_launch(void* const* d_in, const int* in_sizes, int n_in,
                              void* d_out, int out_size, void* d_ws, size_t ws_size,
                              hipStream_t stream) {
    (void)in_sizes; (void)n_in; (void)d_ws; (void)ws_size; (void)out_size;
    const float* x = (const float*)d_in[0];
    // d_in[1] is the fixed (1,3,3,1) outer-product kernel; weights are baked in.
    float* out = (float*)d_out;

    dim3 grid(IH / TROWS, 8 * 256, 1);   // 8 row-tiles x 2048 channels
    upsample2x_1331_kernel<<<grid, dim3(256), 0, stream>>>(x, out);
}